// PointPatchEmbedding_69148973466108
// MI455X (gfx1250) — compile-verified
//
#include <hip/hip_runtime.h>

typedef float v2f __attribute__((ext_vector_type(2)));
typedef float v8f __attribute__((ext_vector_type(8)));

#define B_   8
#define T_   2048
#define N_   64
#define C_   2
#define P_   4
#define E_   256
#define TP_  (T_ / P_)           // 512
#define MROWS  (B_ * TP_ * N_)   // 262144
#define MTILES (MROWS / 16)      // 16384
#define WAVES_PER_BLOCK 8

// D = A(16x4) x B(4x16) + C, f32, two chained WMMAs cover K=8 (j = k*2 + c).
// A layout (ISA 7.12.2): lane L<16 holds M=L, K={0,1}; lane L>=16 holds M=L-16, K={2,3}.
// For our j-ordering, lane L's two A values are points[b, t, n, 0..1] (contiguous float2)
// with t = tp*4 + (L>>4) for WMMA#0 and t+2 for WMMA#1.
__global__ __launch_bounds__(256) void patch_embed_wmma(
    const float* __restrict__ points,   // [B,T,N,C]
    const int*   __restrict__ lengths,  // [B]
    const float* __restrict__ Wt,       // [E,C,P]
    const float* __restrict__ bias,     // [E]
    float* __restrict__ out)            // [MROWS,E] patches ++ [B] patch_lengths (as f32)
{
    const int tid  = threadIdx.x;
    const int lane = tid & 31;
    const int wave = tid >> 5;
    const int mtile = blockIdx.x * WAVES_PER_BLOCK + wave;   // exact cover, no bounds check

    // tiny second output: patch_lengths (stored as float after the patches block)
    if (blockIdx.x == 0 && tid < B_) {
        const int len    = lengths[tid];
        const int padded = ((len + P_ - 1) >> 2) << 2;
        out[(size_t)MROWS * E_ + tid] = (float)(padded >> 2);
    }

    // decode M-tile -> (b, tp, n0); 16 consecutive rows share one (b, tp)
    const int bt = mtile >> 2;
    const int n0 = (mtile & 3) << 4;
    const int b  = bt >> 9;              // / TP_
    const int tp = bt & (TP_ - 1);

    const int len    = lengths[b];
    const int padded = ((len + P_ - 1) >> 2) << 2;

    const int nn = n0 + (lane & 15);     // this lane's M row == point index n
    const int kh = lane >> 4;            // which K-half this lane supplies

    // ---- A: one coalesced float2 per lane per WMMA (each point read once device-wide)
    const int t0 = tp * P_ + kh;         // k = kh      (WMMA #0)
    const int t1 = t0 + 2;               // k = kh + 2  (WMMA #1)
    const int s0 = min(t0, len - 1);     // replicate-last gather
    const int s1 = min(t1, len - 1);
    const float m0 = (t0 < padded) ? 1.0f : 0.0f;
    const float m1 = (t1 < padded) ? 1.0f : 0.0f;

    const v2f* p2 = (const v2f*)points;  // float2 units: [b][t][n]
    v2f a0 = p2[((size_t)b * T_ + s0) * N_ + nn];
    v2f a1 = p2[((size_t)b * T_ + s1) * N_ + nn];
    a0 *= m0;
    a1 *= m1;

    const size_t outRowBase = (size_t)mtile * 16;

    // ---- loop over the 16 E-tiles; A reused 16x from registers
    for (int et = 0; et < 16; ++et) {
        const int e = (et << 4) + (lane & 15);
        const float* we = Wt + e * (C_ * P_);   // W[e, c, k] at e*8 + c*4 + k (8KB, cache-resident)

        v2f b0, b1;
        b0.x = we[kh];        // j = 2*kh + 0 -> (c=0, k=kh)
        b0.y = we[4 + kh];    // j = 2*kh + 1 -> (c=1, k=kh)
        b1.x = we[2 + kh];    // (c=0, k=kh+2)
        b1.y = we[6 + kh];    // (c=1, k=kh+2)

        // init accumulator with bias (all 8 C VGPRs carry the same e for this lane)
        const float bv = bias[e];
        v8f c = { bv, bv, bv, bv, bv, bv, bv, bv };

        c = __builtin_amdgcn_wmma_f32_16x16x4_f32(false, a0, false, b0, (short)0, c, false, false);
        c = __builtin_amdgcn_wmma_f32_16x16x4_f32(false, a1, false, b1, (short)0, c, false, false);

        // D layout: VGPR r -> M = r + 8*(lane>>4), N = lane&15
        // each half-wave store covers a contiguous 64B run of the output row
        #pragma unroll
        for (int r = 0; r < 8; ++r) {
            const size_t row = outRowBase + (size_t)r + ((size_t)kh << 3);
            out[row * E_ + e] = c[r];
        }
    }
}

extern "C" void kernel_launch(void* const* d_in, const int* in_sizes, int n_in,
                              void* d_out, int out_size, void* d_ws, size_t ws_size,
                              hipStream_t stream) {
    const float* points  = (const float*)d_in[0];
    const int*   lengths = (const int*)  d_in[1];
    const float* Wt      = (const float*)d_in[2];
    const float* bias    = (const float*)d_in[3];
    float*       out     = (float*)d_out;

    const dim3 grid(MTILES / WAVES_PER_BLOCK);   // 2048 blocks
    const dim3 block(256);                       // 8 wave32s
    patch_embed_wmma<<<grid, block, 0, stream>>>(points, lengths, Wt, bias, out);
}